// Joint_Latent_43095701848327
// MI455X (gfx1250) — compile-verified
//
#include <hip/hip_runtime.h>
#include <math.h>

typedef __attribute__((ext_vector_type(2))) float v2f;
typedef __attribute__((ext_vector_type(8))) float v8f;

#define ZD 128

// Order-preserving float <-> u32 key (for atomic segment-max on floats that
// may be negative). key(a) < key(b)  <=>  a < b. Init key 0 == "minus inf".
__device__ __forceinline__ unsigned fkey(float f) {
    unsigned b = __float_as_uint(f);
    return (b & 0x80000000u) ? ~b : (b | 0x80000000u);
}
__device__ __forceinline__ float funkey(unsigned k) {
    unsigned b = (k & 0x80000000u) ? (k & 0x7fffffffu) : ~k;
    return __uint_as_float(b);
}

// ---------------------------------------------------------------------------
// 0) Zero-init output (used as the agg accumulator) and per-node scratch.
// ---------------------------------------------------------------------------
__global__ void gat_init(float4* __restrict__ out4, unsigned* __restrict__ m_u,
                         float* __restrict__ denom, unsigned* __restrict__ deg, int n) {
    int i = blockIdx.x * blockDim.x + threadIdx.x;
    if (i < n * (ZD / 4)) out4[i] = make_float4(0.f, 0.f, 0.f, 0.f);
    if (i < n) { m_u[i] = 0u; denom[i] = 0.f; deg[i] = 0u; }
}

// ---------------------------------------------------------------------------
// 1) Projection P[n,2] = Z[n,128] x W[128,2] via V_WMMA_F32_16X16X4_F32.
//    One wave per 16-row tile; K loop of 32 WMMAs.
//    A 16x4 layout: lane m (0..15) rows, half-wave selects K pair (0,1)/(2,3).
//    B 4x16: columns 0/1 carry the two halves of w, rest zero. B is fetched
//    branchlessly from LDS (clamped column + 0/1 mask) so the inner loop is
//    ds_load_b64 -> v_mul -> v_wmma with no EXEC-mask branches.
//    C/D 16x16 f32: VGPR r, lane t -> (M = r + 8*(t>>4), N = t&15).
// ---------------------------------------------------------------------------
__global__ void gat_proj_wmma(const float* __restrict__ z, const float* __restrict__ w,
                              float* __restrict__ ps, float* __restrict__ pd, int n) {
    __shared__ float wlds[2 * ZD];
    for (int i = threadIdx.x; i < 2 * ZD; i += blockDim.x) wlds[i] = w[i];
    __syncthreads();

    int lane = threadIdx.x & 31;
    int tile = blockIdx.x * (blockDim.x >> 5) + (threadIdx.x >> 5);
    if (tile * 16 >= n) return;               // wave-uniform exit: EXEC stays full

    int mrow = lane & 15;
    int hi   = lane >> 4;                     // 0: K=0,1  1: K=2,3
    int row  = tile * 16 + mrow;
    if (row >= n) row = n - 1;                // safe clamp (n % 16 == 0 here)
    const float2* zr = (const float2*)(z + (size_t)row * ZD);
    int ncol = lane & 15;                     // output column this lane owns

    // Branchless B fetch setup: lanes with ncol>=2 read column 0 but mask to 0.
    int   c    = (ncol < 2) ? ncol : 0;
    float msk  = (ncol < 2) ? 1.f : 0.f;
    const float2* wl2 = (const float2*)(wlds + c * ZD);

    v8f acc = {};
    for (int k0 = 0; k0 < ZD; k0 += 4) {
        int   idx = (k0 >> 1) + hi;
        float2 av = zr[idx];
        float2 bv = wl2[idx];
        v2f a; a.x = av.x;       a.y = av.y;
        v2f b; b.x = bv.x * msk; b.y = bv.y * msk;
        acc = __builtin_amdgcn_wmma_f32_16x16x4_f32(
            false, a, false, b, (short)0, acc, false, false);
    }

    if (ncol < 2) {
        float* outp = (ncol == 0) ? ps : pd;
        int base = tile * 16 + 8 * hi;
#pragma unroll
        for (int r = 0; r < 8; ++r)
            if (base + r < n) outp[base + r] = acc[r];
    }
}

// ---------------------------------------------------------------------------
// 2) Per edge: e = selu(ps[src] + pd[dst]); segment-max into m_u; count deg.
// ---------------------------------------------------------------------------
__global__ void gat_edge_max(const float* __restrict__ ps, const float* __restrict__ pd,
                             const int* __restrict__ src, const int* __restrict__ dst,
                             float* __restrict__ e_arr, unsigned* __restrict__ m_u,
                             unsigned* __restrict__ deg, int ne) {
    int i = blockIdx.x * blockDim.x + threadIdx.x;
    if (i >= ne) return;
    int s = src[i], d = dst[i];
    float x = ps[s] + pd[d];
    const float scale = 1.0507009873554805f;
    const float alpha = 1.6732632423543772f;
    float v = (x > 0.f) ? (scale * x) : (scale * alpha * (expf(x) - 1.f));
    e_arr[i] = v;
    atomicMax(&m_u[d], fkey(v));             // global_atomic_max_u32
    atomicAdd(&deg[d], 1u);
}

// ---------------------------------------------------------------------------
// 3) Per edge: e_exp = exp(e - m[dst]); denom[dst] += e_exp (hw f32 atomic).
//    Overwrite e_arr with e_exp for the final pass.
// ---------------------------------------------------------------------------
__global__ void gat_edge_expsum(const int* __restrict__ dst, float* __restrict__ e_arr,
                                const unsigned* __restrict__ m_u,
                                float* __restrict__ denom, int ne) {
    int i = blockIdx.x * blockDim.x + threadIdx.x;
    if (i >= ne) return;
    int d = dst[i];
    float ex = expf(e_arr[i] - funkey(m_u[d]));
    e_arr[i] = ex;
    unsafeAtomicAdd(&denom[d], ex);          // global_atomic_add_f32
}

// ---------------------------------------------------------------------------
// 4) Heavy pass: one wave per edge. Coalesced 512B float4 gather of z[src]
//    (L2-resident: z is 51MB < 192MB L2), scale by a, scatter-add into out.
// ---------------------------------------------------------------------------
__global__ void gat_edge_agg(const float* __restrict__ z, const int* __restrict__ src,
                             const int* __restrict__ dst, const float* __restrict__ e_arr,
                             const float* __restrict__ denom, float* __restrict__ out, int ne) {
    int lane = threadIdx.x & 31;
    int e = blockIdx.x * (blockDim.x >> 5) + (threadIdx.x >> 5);
    if (e >= ne) return;                     // wave-uniform
    int s = src[e], d = dst[e];
    float a = e_arr[e] / denom[d];
    float4 v = ((const float4*)(z + (size_t)s * ZD))[lane];
    float* o = out + (size_t)d * ZD + lane * 4;
    unsafeAtomicAdd(o + 0, a * v.x);
    unsafeAtomicAdd(o + 1, a * v.y);
    unsafeAtomicAdd(o + 2, a * v.z);
    unsafeAtomicAdd(o + 3, a * v.w);
}

// ---------------------------------------------------------------------------
// 5) Nodes with no incoming edges keep their input latent.
// ---------------------------------------------------------------------------
__global__ void gat_finalize(const float4* __restrict__ z4, const unsigned* __restrict__ deg,
                             float4* __restrict__ out4, int n) {
    int i = blockIdx.x * blockDim.x + threadIdx.x;
    if (i >= n * (ZD / 4)) return;
    int node = i >> 5;                       // ZD/4 == 32 chunks per row
    if (deg[node] == 0u) out4[i] = z4[i];
}

extern "C" void kernel_launch(void* const* d_in, const int* in_sizes, int n_in,
                              void* d_out, int out_size, void* d_ws, size_t ws_size,
                              hipStream_t stream) {
    const float* z   = (const float*)d_in[0];
    const float* w   = (const float*)d_in[1];
    const int*   src = (const int*)d_in[2];
    const int*   dst = (const int*)d_in[3];
    int n  = in_sizes[0] / ZD;
    int ne = in_sizes[2];
    float* out = (float*)d_out;

    // Workspace layout (all 4-byte aligned): 5*n + ne floats.
    char* ws = (char*)d_ws;
    float*    ps    = (float*)ws;    ws += (size_t)n  * sizeof(float);
    float*    pd    = (float*)ws;    ws += (size_t)n  * sizeof(float);
    unsigned* m_u   = (unsigned*)ws; ws += (size_t)n  * sizeof(unsigned);
    float*    denom = (float*)ws;    ws += (size_t)n  * sizeof(float);
    unsigned* deg   = (unsigned*)ws; ws += (size_t)n  * sizeof(unsigned);
    float*    e_arr = (float*)ws;

    const int tpb = 256;
    const int wpb = tpb / 32;                // waves per block (wave32)
    int tot4 = n * (ZD / 4);

    gat_init<<<(tot4 + tpb - 1) / tpb, tpb, 0, stream>>>((float4*)out, m_u, denom, deg, n);

    int ntiles = (n + 15) / 16;
    gat_proj_wmma<<<(ntiles + wpb - 1) / wpb, tpb, 0, stream>>>(z, w, ps, pd, n);

    gat_edge_max<<<(ne + tpb - 1) / tpb, tpb, 0, stream>>>(ps, pd, src, dst, e_arr, m_u, deg, ne);
    gat_edge_expsum<<<(ne + tpb - 1) / tpb, tpb, 0, stream>>>(dst, e_arr, m_u, denom, ne);
    gat_edge_agg<<<(ne + wpb - 1) / wpb, tpb, 0, stream>>>(z, src, dst, e_arr, denom, out, ne);
    gat_finalize<<<(tot4 + tpb - 1) / tpb, tpb, 0, stream>>>((const float4*)z, deg, (float4*)out, n);
}